// SMFNet_3375844295380
// MI455X (gfx1250) — compile-verified
//
#include <hip/hip_runtime.h>
#include <hip/hip_bf16.h>

// ---------------------------------------------------------------------------
// SMFNet on MI455X (gfx1250, wave32).
//   V = relu(relu(X g0^T + b0) g1^T + b1)
//   per layer m: F = relu(X f0^T + f0b); scores only on chord (diag + superdiag)
//   V'[i] = s0[i] V[i] + s1[i] V[(i+1)%N]
// Dense 64-K GEMMs run on V_WMMA_F32_16X16X4_F32 (fp32 in, fp32 acc).
// ---------------------------------------------------------------------------

#define SEQ_N 8192
#define DIM   64

typedef __attribute__((ext_vector_type(2))) float v2f;
typedef __attribute__((ext_vector_type(8))) float v8f;

// O[N,64] = relu(A[N,64] @ W[64,64]^T + bias[64])
// One wave per 16-row tile; 8 waves (256 thr) per block -> 128 rows/block.
__global__ __launch_bounds__(256) void gemm64_relu_kernel(
    const float* __restrict__ A,
    const float* __restrict__ W,
    const float* __restrict__ bias,
    float* __restrict__ O)
{
    const int wave = threadIdx.x >> 5;
    const int lane = threadIdx.x & 31;
    const int lm   = lane & 15;   // 0..15
    const int lh   = lane >> 4;   // 0..1
    const int row0 = (blockIdx.x * 8 + wave) * 16;

    const float* arow = A + (size_t)(row0 + lm) * DIM + 2 * lh;

    v8f acc0 = {}, acc1 = {}, acc2 = {}, acc3 = {};

    #pragma unroll
    for (int kk = 0; kk < DIM; kk += 4) {
        // A tile 16x4: lanes lh=0 hold K=kk,kk+1 ; lanes lh=1 hold K=kk+2,kk+3
        v2f a = *(const v2f*)(arow + kk);
        // B tile 4x16 for n-tile nt: B[k][n] = W[n][k] -> contiguous float2 in W
        v2f b0 = *(const v2f*)(W + (size_t)( 0 + lm) * DIM + kk + 2 * lh);
        v2f b1 = *(const v2f*)(W + (size_t)(16 + lm) * DIM + kk + 2 * lh);
        v2f b2 = *(const v2f*)(W + (size_t)(32 + lm) * DIM + kk + 2 * lh);
        v2f b3 = *(const v2f*)(W + (size_t)(48 + lm) * DIM + kk + 2 * lh);
        acc0 = __builtin_amdgcn_wmma_f32_16x16x4_f32(false, a, false, b0, (short)0, acc0, false, false);
        acc1 = __builtin_amdgcn_wmma_f32_16x16x4_f32(false, a, false, b1, (short)0, acc1, false, false);
        acc2 = __builtin_amdgcn_wmma_f32_16x16x4_f32(false, a, false, b2, (short)0, acc2, false, false);
        acc3 = __builtin_amdgcn_wmma_f32_16x16x4_f32(false, a, false, b3, (short)0, acc3, false, false);
    }

    // C/D layout: VGPR j -> M = j + 8*lh, N = nt*16 + lm
    v8f accs[4] = {acc0, acc1, acc2, acc3};
    #pragma unroll
    for (int nt = 0; nt < 4; ++nt) {
        const int col = nt * 16 + lm;
        const float bv = bias[col];
        #pragma unroll
        for (int j = 0; j < 8; ++j) {
            const int m = j + 8 * lh;
            float v = accs[nt][j] + bv;
            O[(size_t)(row0 + m) * DIM + col] = v > 0.f ? v : 0.f;
        }
    }
}

// Chord-sparse score + apply, one wave per row i:
//   s0 = relu(<F[i],WL[i]> + bL[i]); s1 = relu(<F[i],WL[i+1]> + bL[i+1])
//   Vout[i] = s0*Vin[i] + s1*Vin[(i+1)%N]
__global__ __launch_bounds__(256) void chord_apply_kernel(
    const float* __restrict__ F,
    const float* __restrict__ WL,
    const float* __restrict__ bL,
    const float* __restrict__ Vin,
    float* __restrict__ Vout)
{
    const int wave = threadIdx.x >> 5;
    const int lane = threadIdx.x & 31;
    const int i   = blockIdx.x * 8 + wave;
    const int ip1 = (i + 1) & (SEQ_N - 1);

    const v2f f  = *(const v2f*)(F  + (size_t)i   * DIM + lane * 2);
    const v2f w0 = *(const v2f*)(WL + (size_t)i   * DIM + lane * 2);
    const v2f w1 = *(const v2f*)(WL + (size_t)ip1 * DIM + lane * 2);

    float d0 = f.x * w0.x + f.y * w0.y;
    float d1 = f.x * w1.x + f.y * w1.y;
    // wave32 butterfly reduction
    #pragma unroll
    for (int off = 16; off > 0; off >>= 1) {
        d0 += __shfl_xor(d0, off, 32);
        d1 += __shfl_xor(d1, off, 32);
    }
    float s0 = d0 + bL[i];   s0 = s0 > 0.f ? s0 : 0.f;
    float s1 = d1 + bL[ip1]; s1 = s1 > 0.f ? s1 : 0.f;

    const v2f vi = *(const v2f*)(Vin + (size_t)i   * DIM + lane * 2);
    const v2f vj = *(const v2f*)(Vin + (size_t)ip1 * DIM + lane * 2);
    v2f o;
    o.x = s0 * vi.x + s1 * vj.x;
    o.y = s0 * vi.y + s1 * vj.y;
    *(v2f*)(Vout + (size_t)i * DIM + lane * 2) = o;
}

extern "C" void kernel_launch(void* const* d_in, const int* in_sizes, int n_in,
                              void* d_out, int out_size, void* d_ws, size_t ws_size,
                              hipStream_t stream)
{
    // setup_inputs() order: X, g_w, g_b, f0_w, f0_b, fL_w, fL_b (all float32)
    const float* X    = (const float*)d_in[0];   // [N,64]
    const float* g_w  = (const float*)d_in[1];   // [2,64,64]
    const float* g_b  = (const float*)d_in[2];   // [2,64]
    const float* f0_w = (const float*)d_in[3];   // [2,64,64]
    const float* f0_b = (const float*)d_in[4];   // [2,64]
    const float* fL_w = (const float*)d_in[5];   // [2,N,64]
    const float* fL_b = (const float*)d_in[6];   // [2,N]
    float* out = (float*)d_out;                  // [N,64]

    const size_t mat = (size_t)SEQ_N * DIM;      // 524288 floats
    float* T1 = (float*)d_ws;                    // relu(X g0^T + b0)
    float* V0 = T1 + mat;                        // after g-stack
    float* V1 = V0 + mat;                        // after chord layer 0
    float* Fb = V1 + mat;                        // f0 activations (reused)
    (void)in_sizes; (void)n_in; (void)out_size; (void)ws_size;

    const dim3 gblk(SEQ_N / 128);   // 64 blocks, 8 waves x 16 rows each
    const dim3 cblk(SEQ_N / 8);     // 1024 blocks, wave-per-row
    const dim3 thr(256);

    // g stack
    gemm64_relu_kernel<<<gblk, thr, 0, stream>>>(X,  g_w,             g_b,       T1);
    gemm64_relu_kernel<<<gblk, thr, 0, stream>>>(T1, g_w + DIM * DIM, g_b + DIM, V0);
    // layer 0
    gemm64_relu_kernel<<<gblk, thr, 0, stream>>>(X, f0_w,             f0_b,       Fb);
    chord_apply_kernel<<<cblk, thr, 0, stream>>>(Fb, fL_w,            fL_b,       V0, V1);
    // layer 1
    gemm64_relu_kernel<<<gblk, thr, 0, stream>>>(X, f0_w + DIM * DIM, f0_b + DIM, Fb);
    chord_apply_kernel<<<cblk, thr, 0, stream>>>(Fb, fL_w + mat,      fL_b + SEQ_N, V1, out);
}